// SoftSW_82386062672159
// MI455X (gfx1250) — compile-verified
//
#include <hip/hip_runtime.h>

#define NEGV -1.0e30f

typedef float v2f __attribute__((ext_vector_type(2)));
typedef float v8f __attribute__((ext_vector_type(8)));

__device__ __forceinline__ float lse2(float a, float b) {
    float m = fmaxf(a, b);
    return m + __logf(__expf(a - m) + __expf(b - m));
}
__device__ __forceinline__ float lse3(float a, float b, float c) {
    float m = fmaxf(fmaxf(a, b), c);
    return m + __logf(__expf(a - m) + __expf(b - m) + __expf(c - m));
}
__device__ __forceinline__ float lse4z(float b, float c, float d) {
    // logsumexp(0, b, c, d)
    float m = fmaxf(fmaxf(0.0f, b), fmaxf(c, d));
    return m + __logf(__expf(0.0f - m) + __expf(b - m) + __expf(c - m) + __expf(d - m));
}

// ---------------------------------------------------------------------------
// Kernel 1: row norms  x2[b,n] = sum_d x^2,  y2[b,m] = sum_d y^2
// ---------------------------------------------------------------------------
__global__ void norms_kernel(const float* __restrict__ x, const float* __restrict__ y,
                             float* __restrict__ x2, float* __restrict__ y2) {
    int idx = blockIdx.x * blockDim.x + threadIdx.x;   // 0 .. 4095
    const int BN = 4 * 512;
    const float* src;
    float* dst;
    if (idx < BN) { src = x + (size_t)idx * 128; dst = x2 + idx; }
    else          { int j = idx - BN; src = y + (size_t)j * 128; dst = y2 + j; }
    const float4* s4 = (const float4*)src;
    float acc = 0.0f;
    #pragma unroll
    for (int k = 0; k < 32; ++k) {
        float4 v = s4[k];
        acc += v.x * v.x + v.y * v.y + v.z * v.z + v.w * v.w;
    }
    *dst = acc;
}

// ---------------------------------------------------------------------------
// Kernel 2: S[b,n,m] = x2 + y2 - 2 * x . y  via V_WMMA_F32_16X16X4_F32
// One wave per 16x16 output tile; 32 WMMAs over K=128.
// A 16x4 f32 layout: lanes 0-15 -> M=lane, VGPR0/1 = K0/K1; lanes 16-31 -> K2/K3.
// B 4x16 f32 layout mirrors it (row of y = column of B).
// ---------------------------------------------------------------------------
__global__ void __launch_bounds__(256) gemm_kernel(const float* __restrict__ x,
                                                   const float* __restrict__ y,
                                                   const float* __restrict__ x2,
                                                   const float* __restrict__ y2,
                                                   float* __restrict__ S) {
    const int wave = threadIdx.x >> 5;
    const int tile = blockIdx.x * 8 + wave;       // 0 .. 4095 = B(4) * 32 * 32
    const int b    = tile >> 10;
    const int tr   = (tile >> 5) & 31;
    const int tc   = tile & 31;
    const int lane = threadIdx.x & 31;
    const int half = lane >> 4;
    const int l16  = lane & 15;
    const int row0 = tr * 16, col0 = tc * 16;

    const float* Ap = x + ((size_t)b * 512 + row0 + l16) * 128 + 2 * half;
    const float* Bp = y + ((size_t)b * 512 + col0 + l16) * 128 + 2 * half;

    v8f acc = {0.f, 0.f, 0.f, 0.f, 0.f, 0.f, 0.f, 0.f};
    #pragma unroll
    for (int k = 0; k < 128; k += 4) {
        v2f a  = *(const v2f*)(Ap + k);
        v2f bb = *(const v2f*)(Bp + k);
        acc = __builtin_amdgcn_wmma_f32_16x16x4_f32(false, a, false, bb,
                                                    (short)0, acc, false, false);
    }

    float yc = y2[b * 512 + col0 + l16];
    float* Srow = S + ((size_t)b * 512 + row0) * 512;
    #pragma unroll
    for (int r = 0; r < 8; ++r) {
        int grow = r + 8 * half;                  // C/D layout: lanes>=16 hold M=r+8
        float sv = x2[b * 512 + row0 + grow] + yc - 2.0f * acc[r];
        Srow[(size_t)grow * 512 + col0 + l16] = sv;
    }
}

// ---------------------------------------------------------------------------
// Kernel 3: soft Smith-Waterman DP, one block (512 threads = 16 waves) per batch.
// Row-parallel d/iy; Ix row-recurrence solved with a log-space associative scan.
// ---------------------------------------------------------------------------
__global__ void __launch_bounds__(512) sw_kernel(const float* __restrict__ S,
                                                 const float* __restrict__ go,
                                                 const float* __restrict__ ge,
                                                 float* __restrict__ probas,
                                                 float* __restrict__ vals) {
    const int b    = blockIdx.x;
    const int tid  = threadIdx.x;                 // column j-1  (0..511)
    const int lane = tid & 31;
    const int wid  = tid >> 5;

    __shared__ float sD[513], sIx[513], sIy[513];
    __shared__ float sWa[16], sWb[16];
    __shared__ float sRm[16], sRs[16];

    for (int k = tid; k < 513; k += 512) { sD[k] = NEGV; sIx[k] = NEGV; sIy[k] = NEGV; }
    __syncthreads();

    const float* Sb = S + (size_t)b * 512 * 512;
    float m_acc = NEGV, s_acc = 0.0f;

    for (int i = 0; i < 512; ++i) {
        float sv  = Sb[(size_t)i * 512 + tid];
        float g_o = go[i * 512 + tid];
        float g_e = ge[i * 512 + tid];
        if (i + 1 < 512) {                        // hide next row behind the scan
            __builtin_prefetch(&Sb[(size_t)(i + 1) * 512 + tid], 0, 1);
            __builtin_prefetch(&go[(i + 1) * 512 + tid], 0, 1);
            __builtin_prefetch(&ge[(i + 1) * 512 + tid], 0, 1);
        }
        float Dd = sD[tid],     Ixd = sIx[tid],     Iyd = sIy[tid];      // diagonal
        float Du = sD[tid + 1], Ixu = sIx[tid + 1], Iyu = sIy[tid + 1];  // up
        __syncthreads();                                                 // reads done

        float d  = sv + lse4z(Dd, Ixd, Iyd);
        float iy = lse3(Du - g_o, Ixu - g_o, Iyu - g_e);
        sD[tid + 1] = d;
        sIy[tid + 1] = iy;
        __syncthreads();                                                 // new d visible

        // Ix scan: op_j : x -> lse2(x - ge_j, d_{j-1} - go_j)
        float d_left = sD[tid];                   // sD[0] is the NEG pad
        float a  = -g_e;
        float bo = d_left - g_o;
        #pragma unroll
        for (int off = 1; off < 32; off <<= 1) {  // intra-wave inclusive scan
            float pa = __shfl_up(a, off);
            float pb = __shfl_up(bo, off);
            if (lane >= off) { bo = lse2(a + pb, bo); a += pa; }
        }
        if (lane == 31) { sWa[wid] = a; sWb[wid] = bo; }
        __syncthreads();

        if (wid == 0) {                           // scan the 16 wave totals
            float wa = (lane < 16) ? sWa[lane] : 0.0f;
            float wb = (lane < 16) ? sWb[lane] : NEGV;
            #pragma unroll
            for (int off = 1; off < 16; off <<= 1) {
                float pa = __shfl_up(wa, off);
                float pb = __shfl_up(wb, off);
                if (lane >= off) { wb = lse2(wa + pb, wb); wa += pa; }
            }
            if (lane < 16) { sWa[lane] = wa; sWb[lane] = wb; }
        }
        __syncthreads();

        float ix = (wid > 0) ? lse2(a + sWb[wid - 1], bo) : bo;   // x0 = NEG drops out
        sIx[tid + 1] = ix;

        // streaming logsumexp of d (pads underflow to exactly 0 contribution)
        if (d <= m_acc) { s_acc += __expf(d - m_acc); }
        else            { s_acc = s_acc * __expf(m_acc - d) + 1.0f; m_acc = d; }

        probas[((size_t)b * 512 + i) * 512 + tid] = d;
        __syncthreads();                                                 // ix visible
    }

    // block-wide logsumexp reduction of (m, s) pairs
    #pragma unroll
    for (int off = 16; off > 0; off >>= 1) {
        float mo = __shfl_down(m_acc, off);
        float so = __shfl_down(s_acc, off);
        float M  = fmaxf(m_acc, mo);
        s_acc = s_acc * __expf(m_acc - M) + so * __expf(mo - M);
        m_acc = M;
    }
    if (lane == 0) { sRm[wid] = m_acc; sRs[wid] = s_acc; }
    __syncthreads();
    if (tid == 0) {
        float M = NEGV, Ss = 0.0f;
        for (int w = 0; w < 16; ++w) {
            float m2 = sRm[w], s2 = sRs[w];
            float Mn = fmaxf(M, m2);
            Ss = Ss * __expf(M - Mn) + s2 * __expf(m2 - Mn);
            M = Mn;
        }
        vals[b] = M + __logf(Ss);
    }
}

// ---------------------------------------------------------------------------
// Kernel 4: loss = mean of per-batch values
// ---------------------------------------------------------------------------
__global__ void loss_kernel(const float* __restrict__ vals, float* __restrict__ out) {
    out[0] = (vals[0] + vals[1] + vals[2] + vals[3]) * 0.25f;
}

extern "C" void kernel_launch(void* const* d_in, const int* in_sizes, int n_in,
                              void* d_out, int out_size, void* d_ws, size_t ws_size,
                              hipStream_t stream) {
    const float* x  = (const float*)d_in[0];   // (4,512,128)
    const float* y  = (const float*)d_in[1];   // (4,512,128)
    const float* go = (const float*)d_in[2];   // (512,512)
    const float* ge = (const float*)d_in[3];   // (512,512)
    float* out = (float*)d_out;                // [loss, probas(4,512,512)]

    float* S    = (float*)d_ws;                // 4*512*512 floats (4 MB)
    float* x2   = S + (size_t)4 * 512 * 512;   // 2048
    float* y2   = x2 + 4 * 512;                // 2048
    float* vals = y2 + 4 * 512;                // 4

    norms_kernel<<<16, 256, 0, stream>>>(x, y, x2, y2);
    gemm_kernel<<<512, 256, 0, stream>>>(x, y, x2, y2, S);
    sw_kernel<<<4, 512, 0, stream>>>(S, go, ge, out + 1, vals);
    loss_kernel<<<1, 1, 0, stream>>>(vals, out);
}